// PositionClassifier_30081950941187
// MI455X (gfx1250) — compile-verified
//
#include <hip/hip_runtime.h>
#include <cstdint>

#define NNODES 100000
#define NEDGES 1000000
#define CH     128
#define NCLS   3

typedef __attribute__((ext_vector_type(2))) float v2f;
typedef __attribute__((ext_vector_type(8))) float v8f;

// ---------------------------------------------------------------------------
// Edge scatter-add: one wave32 per edge. Each lane moves 4 channels (float4
// coalesced load from the source row, 4 f32 atomics into the accumulator).
// Lane 0 bumps the degree counter. Accumulator (51MB) lives in the 192MB L2,
// so the atomic RMW traffic never reaches HBM.
// ---------------------------------------------------------------------------
__global__ __launch_bounds__(256)
void sage_scatter_accum(const float* __restrict__ x,
                        const long long* __restrict__ src,
                        const long long* __restrict__ dst,
                        float* __restrict__ agg,
                        float* __restrict__ cnt) {
    const int edge = (int)((blockIdx.x * blockDim.x + threadIdx.x) >> 5);
    const int lane = threadIdx.x & 31;
    if (edge >= NEDGES) return;
    const long long s = src[edge];
    const long long d = dst[edge];
    const float4 v = *((const float4*)(x + s * CH) + lane);
    float* ap = agg + d * CH + lane * 4;
    atomicAdd(ap + 0, v.x);
    atomicAdd(ap + 1, v.y);
    atomicAdd(ap + 2, v.z);
    atomicAdd(ap + 3, v.w);
    if (lane == 0) atomicAdd(cnt + d, 1.0f);
}

// mean = agg / max(cnt, 1)
__global__ __launch_bounds__(256)
void sage_mean_div(float* __restrict__ agg, const float* __restrict__ cnt) {
    const int i = blockIdx.x * blockDim.x + threadIdx.x;   // over NNODES*CH
    const float c = cnt[i >> 7];                            // node = i / 128
    agg[i] = agg[i] / fmaxf(c, 1.0f);
}

// ---------------------------------------------------------------------------
// Fused SAGE linear: out = relu( mean @ Wl + bl + xroot @ Wr )
// One block = 16 output rows; 8 waves, each wave owns a 16-wide column tile.
// v_wmma_f32_16x16x4_f32 accumulator chain over K=128 for both weight
// matrices (64 WMMAs per tile), then bias + ReLU on the D layout.
//
// ISA VGPR layouts (CDNA5 ISA):
//   A 16x4 f32 : lanes 0-15 -> M=lane, VGPR{0,1}=K{0,1}; lanes 16-31 -> K{2,3}
//   B 4x16 f32 : lanes 0-15 -> N=lane, VGPR{0,1}=K{0,1}; lanes 16-31 -> K{2,3}
//   C/D 16x16  : VGPR r -> M = r + 8*(lane>=16), N = lane&15 (tile-local)
// ---------------------------------------------------------------------------
__global__ __launch_bounds__(256)
void sage_gemm_relu(const float* __restrict__ Amean,
                    const float* __restrict__ Aroot,
                    const float* __restrict__ Wl,
                    const float* __restrict__ bl,
                    const float* __restrict__ Wr,
                    float* __restrict__ Out) {
    const int rowbase = blockIdx.x * 16;
    const int waveId  = threadIdx.x >> 5;       // 0..7 -> 16-col tile
    const int lane    = threadIdx.x & 31;
    const int half    = lane >> 4;              // 0 or 1
    const int ml      = lane & 15;
    const int n       = waveId * 16 + ml;       // B/N column for this lane
    const int k_half  = 2 * half;               // K offset 0 or 2

    v8f c = {};

    const float* arow = Amean + (long)(rowbase + ml) * CH;
    for (int k0 = 0; k0 < CH; k0 += 4) {
        v2f a = *(const v2f*)(arow + k0 + k_half);
        v2f b;
        b.x = Wl[(k0 + k_half) * CH + n];
        b.y = Wl[(k0 + k_half + 1) * CH + n];
        c = __builtin_amdgcn_wmma_f32_16x16x4_f32(false, a, false, b,
                                                  (short)0, c, false, false);
    }
    const float* xrow = Aroot + (long)(rowbase + ml) * CH;
    for (int k0 = 0; k0 < CH; k0 += 4) {
        v2f a = *(const v2f*)(xrow + k0 + k_half);
        v2f b;
        b.x = Wr[(k0 + k_half) * CH + n];
        b.y = Wr[(k0 + k_half + 1) * CH + n];
        c = __builtin_amdgcn_wmma_f32_16x16x4_f32(false, a, false, b,
                                                  (short)0, c, false, false);
    }

    // Write D with bias + ReLU. Lane's column == n; rows rowbase + 8*half + r.
    const float bias = bl[n];
    const int r0 = rowbase + half * 8;
#pragma unroll
    for (int r = 0; r < 8; ++r) {
        float v = c[r] + bias;
        Out[(long)(r0 + r) * CH + n] = v > 0.0f ? v : 0.0f;
    }
}

// logits = h @ w_cls + b_cls   (128 x 3, bandwidth-trivial)
__global__ __launch_bounds__(256)
void sage_classifier(const float* __restrict__ h,
                     const float* __restrict__ wc,
                     const float* __restrict__ bc,
                     float* __restrict__ out) {
    const int node = blockIdx.x * blockDim.x + threadIdx.x;
    if (node >= NNODES) return;
    float a0 = bc[0], a1 = bc[1], a2 = bc[2];
    const float* row = h + (long)node * CH;
#pragma unroll 4
    for (int k = 0; k < CH; ++k) {
        const float v = row[k];
        a0 = fmaf(v, wc[k * NCLS + 0], a0);
        a1 = fmaf(v, wc[k * NCLS + 1], a1);
        a2 = fmaf(v, wc[k * NCLS + 2], a2);
    }
    out[node * NCLS + 0] = a0;
    out[node * NCLS + 1] = a1;
    out[node * NCLS + 2] = a2;
}

extern "C" void kernel_launch(void* const* d_in, const int* in_sizes, int n_in,
                              void* d_out, int out_size, void* d_ws, size_t ws_size,
                              hipStream_t stream) {
    const float*     x    = (const float*)d_in[0];
    const long long* eidx = (const long long*)d_in[1];   // int64 (2, NEDGES)
    const float*     w_l1 = (const float*)d_in[2];
    const float*     b_l1 = (const float*)d_in[3];
    const float*     w_r1 = (const float*)d_in[4];
    const float*     w_l2 = (const float*)d_in[5];
    const float*     b_l2 = (const float*)d_in[6];
    const float*     w_r2 = (const float*)d_in[7];
    const float*     w_cls = (const float*)d_in[8];
    const float*     b_cls = (const float*)d_in[9];
    float* out = (float*)d_out;

    const long long* src = eidx;            // row 0
    const long long* dst = eidx + NEDGES;   // row 1

    const size_t NM = (size_t)NNODES * CH;  // 12.8M floats
    float* agg = (float*)d_ws;              // [NM]  (reused both layers)
    float* cnt = agg + NM;                  // [NNODES]
    float* h1  = cnt + NNODES;              // [NM]
    float* h2  = h1 + NM;                   // [NM]  total ~154 MB

    const int scatterBlocks = NEDGES / 8;           // 8 edges (waves) / block
    const int meanBlocks    = (int)(NM / 256);
    const int gemmBlocks    = NNODES / 16;          // 6250, exact
    const int clsBlocks     = (NNODES + 255) / 256;

    // ---- layer 1 ----
    hipMemsetAsync(agg, 0, (NM + NNODES) * sizeof(float), stream);
    sage_scatter_accum<<<scatterBlocks, 256, 0, stream>>>(x, src, dst, agg, cnt);
    sage_mean_div<<<meanBlocks, 256, 0, stream>>>(agg, cnt);
    sage_gemm_relu<<<gemmBlocks, 256, 0, stream>>>(agg, x, w_l1, b_l1, w_r1, h1);

    // ---- layer 2 ----
    hipMemsetAsync(agg, 0, (NM + NNODES) * sizeof(float), stream);
    sage_scatter_accum<<<scatterBlocks, 256, 0, stream>>>(h1, src, dst, agg, cnt);
    sage_mean_div<<<meanBlocks, 256, 0, stream>>>(agg, cnt);
    sage_gemm_relu<<<gemmBlocks, 256, 0, stream>>>(agg, h1, w_l2, b_l2, w_r2, h2);

    // ---- classifier ----
    sage_classifier<<<clsBlocks, 256, 0, stream>>>(h2, w_cls, b_cls, out);
}